// DualTowerModel_28862180229803
// MI455X (gfx1250) — compile-verified
//
#include <hip/hip_runtime.h>
#include <hip/hip_bf16.h>
#include <math.h>

// ---------------- model constants (fixed by the reference) ----------------
#define DIM   64
#define HLEN  200
#define NGEN  6
#define LAM   0.001f

typedef __attribute__((ext_vector_type(2))) float v2f;
typedef __attribute__((ext_vector_type(8))) float v8f;

__device__ __forceinline__ float wave_sum(float v) {
#pragma unroll
    for (int off = 16; off > 0; off >>= 1)
        v += __shfl_xor(v, off, 32);
    return v;
}

// =========================================================================
// Kernel 1: embedding gathers + weighted pooling + continuous projections.
// 256 threads = 8 waves; one wave32 per batch row.
// History gather (dominant ~840MB path): weights staged in LDS once per
// row, then 2 items/iteration with one global_load_b128 per lane
// (16 lanes x float4 = one 256B embedding row per half-wave).
// Builds u_cat [B,256] = [u_e | hist_pool | ug_pool | u_cont]
//        i_cat [B,192] = [i_e | ig_pool | i_cont]
// =========================================================================
__global__ __launch_bounds__(256) void pool_features_kernel(
    const int* __restrict__ user_id, const int* __restrict__ history,
    const int* __restrict__ top_genres, const int* __restrict__ item_id,
    const int* __restrict__ tmdb_genres, const float* __restrict__ ts_diff,
    const float* __restrict__ u_avg, const float* __restrict__ activity,
    const float* __restrict__ rel_year, const float* __restrict__ i_avg,
    const float* __restrict__ revenue,
    const float* __restrict__ emb_item, const float* __restrict__ emb_genre,
    const float* __restrict__ emb_user,
    const float* __restrict__ ut_cont_w, const float* __restrict__ ut_cont_b,
    const float* __restrict__ it_cont_w, const float* __restrict__ it_cont_b,
    float* __restrict__ u_cat, float* __restrict__ i_cat, int B)
{
    __shared__ int   s_idx[8][HLEN];
    __shared__ float s_w[8][HLEN];

    const int wave = threadIdx.x >> 5;   // row within block
    const int lane = threadIdx.x & 31;
    const int row  = blockIdx.x * 8 + wave;
    const bool live = (row < B);

    // ---- stage 1: per-row decay weights into LDS (coalesced, 1x exp per slot)
    float wpart = 0.f;
    if (live) {
        const int*   hrow = history + (size_t)row * HLEN;
        const float* trow = ts_diff + (size_t)row * HLEN;
        for (int l = lane; l < HLEN; l += 32) {
            const int   idx = hrow[l];
            const float w   = (idx > 0) ? expf(-LAM * trow[l]) : 0.f;
            s_idx[wave][l] = idx;
            s_w[wave][l]   = w;
            wpart += w;
        }
    }
    __syncthreads();   // cross-lane LDS dependence: do not reorder

    if (!live) return;

    const float hwsum = wave_sum(wpart);

    // ---- stage 2: history gather, 2 items/iter, b128 per lane ----
    const int c    = lane & 15;    // float4 chunk: dims [4c, 4c+4)
    const int half = lane >> 4;    // which of the 2 items this half handles
    float4 acc = make_float4(0.f, 0.f, 0.f, 0.f);
    for (int l = 0; l < HLEN; l += 2) {
        const int   idx = s_idx[wave][l + half];
        const float w   = s_w[wave][l + half];
        const float4 e  = *(const float4*)(emb_item + (size_t)idx * DIM + 4 * c);
        acc.x += w * e.x; acc.y += w * e.y; acc.z += w * e.z; acc.w += w * e.w;
    }
    // combine the two halves (dims match across halves: lane&15)
    acc.x += __shfl_xor(acc.x, 16, 32);
    acc.y += __shfl_xor(acc.y, 16, 32);
    acc.z += __shfl_xor(acc.z, 16, 32);
    acc.w += __shfl_xor(acc.w, 16, 32);

    float* uo = u_cat + (size_t)row * 256;
    float* io = i_cat + (size_t)row * 192;

    if (half == 0) {
        const float inv = 1.f / (hwsum + 1e-8f);
        float4 hp = make_float4(acc.x * inv, acc.y * inv, acc.z * inv, acc.w * inv);
        *(float4*)(uo + 64 + 4 * c) = hp;
    }

    // ---- remaining features: float2 per lane (dims [2*lane, 2*lane+2)) ----
    const int d0 = 2 * lane;

    // user embedding
    {
        const float2 ue = *(const float2*)(emb_user + (size_t)user_id[row] * DIM + d0);
        *(float2*)(uo + 0 + d0) = ue;
    }
    // user genre pooling (mask-weighted mean, 6 items)
    {
        const int* grow = top_genres + (size_t)row * NGEN;
        float2 ga = make_float2(0.f, 0.f);
        float gw = 0.f;
#pragma unroll
        for (int g = 0; g < NGEN; ++g) {
            const int   idx = grow[g];
            const float m   = (idx > 0) ? 1.f : 0.f;
            const float2 e  = *(const float2*)(emb_genre + (size_t)idx * DIM + d0);
            ga.x += m * e.x; ga.y += m * e.y; gw += m;
        }
        const float inv = 1.f / (gw + 1e-8f);
        *(float2*)(uo + 128 + d0) = make_float2(ga.x * inv, ga.y * inv);
    }
    // user continuous: relu([u_avg, act] @ W.T + b), W is [64,2] -> float4/lane
    {
        const float ua = u_avg[row], ac = activity[row];
        const float4 w4 = *(const float4*)(ut_cont_w + 2 * d0); // w[d0][0..1], w[d0+1][0..1]
        const float2 b2 = *(const float2*)(ut_cont_b + d0);
        float2 uc;
        uc.x = fmaxf(ua * w4.x + ac * w4.y + b2.x, 0.f);
        uc.y = fmaxf(ua * w4.z + ac * w4.w + b2.y, 0.f);
        *(float2*)(uo + 192 + d0) = uc;
    }

    // item embedding
    {
        const float2 ie = *(const float2*)(emb_item + (size_t)item_id[row] * DIM + d0);
        *(float2*)(io + 0 + d0) = ie;
    }
    // item genre pooling
    {
        const int* grow = tmdb_genres + (size_t)row * NGEN;
        float2 ga = make_float2(0.f, 0.f);
        float gw = 0.f;
#pragma unroll
        for (int g = 0; g < NGEN; ++g) {
            const int   idx = grow[g];
            const float m   = (idx > 0) ? 1.f : 0.f;
            const float2 e  = *(const float2*)(emb_genre + (size_t)idx * DIM + d0);
            ga.x += m * e.x; ga.y += m * e.y; gw += m;
        }
        const float inv = 1.f / (gw + 1e-8f);
        *(float2*)(io + 64 + d0) = make_float2(ga.x * inv, ga.y * inv);
    }
    // item continuous: relu([year, i_avg, rev] @ W.T + b), W is [64,3]
    {
        const float ry = rel_year[row], ia = i_avg[row], rv = revenue[row];
        float2 ic;
        ic.x = fmaxf(ry * it_cont_w[d0 * 3 + 0] + ia * it_cont_w[d0 * 3 + 1] +
                     rv * it_cont_w[d0 * 3 + 2] + it_cont_b[d0], 0.f);
        ic.y = fmaxf(ry * it_cont_w[d0 * 3 + 3] + ia * it_cont_w[d0 * 3 + 4] +
                     rv * it_cont_w[d0 * 3 + 5] + it_cont_b[d0 + 1], 0.f);
        *(float2*)(io + 128 + d0) = ic;
    }
}

// =========================================================================
// Kernels 2/3: tower MLP with f32 WMMA (V_WMMA_F32_16X16X4_F32).
//   X [B,K1] -> h = relu(LN(X @ W1.T + b1)) [B,128] -> l2norm(h @ W2.T + b2)
// 256 threads / block = 8 wave32. Each block owns 16 batch rows.
// =========================================================================
template <int K1>
__global__ __launch_bounds__(256) void tower_mlp_kernel(
    const float* __restrict__ X,   // [B, K1]
    const float* __restrict__ W1,  // [128, K1] row-major (h = X @ W1.T)
    const float* __restrict__ b1,  // [128]
    const float* __restrict__ gam, // [128]
    const float* __restrict__ bet, // [128]
    const float* __restrict__ W2,  // [64, 128] row-major
    const float* __restrict__ b2,  // [64]
    float* __restrict__ out)       // [B, 64]
{
    __shared__ float h[16 * 128];
    __shared__ float o[16 * 64];
    __shared__ float rmean[16], rrstd[16], rnorm[16];

    const int tid  = threadIdx.x;
    const int wave = tid >> 5;     // wave32
    const int lane = tid & 31;
    const int row0 = blockIdx.x * 16;

    // ---- GEMM1: per-wave 16x16 tile, K-loop in steps of 4 ----
    {
        const int nbase = wave * 16;
        const int mn    = lane & 15;        // A row index / B col index
        const int ks    = (lane >> 4) * 2;  // K sub-pair per A/B layout
        v8f acc = {};
        const float* xrow = X  + (size_t)(row0 + mn) * K1;
        const float* wrow = W1 + (size_t)(nbase + mn) * K1;
        for (int k = 0; k < K1; k += 4) {
            v2f a, b;
            a.x = xrow[k + ks]; a.y = xrow[k + ks + 1];
            b.x = wrow[k + ks]; b.y = wrow[k + ks + 1];
            acc = __builtin_amdgcn_wmma_f32_16x16x4_f32(
                false, a, false, b, (short)0, acc, false, false);
        }
        // C/D layout: VGPR r -> M = r + 8*(lane>>4), N = lane&15
        const int mr = (lane >> 4) * 8;
        const int nc = nbase + (lane & 15);
#pragma unroll
        for (int r = 0; r < 8; ++r)
            h[(mr + r) * 128 + nc] = acc[r] + b1[nc];
    }
    __syncthreads();

    // ---- LayerNorm row stats over 128 cols ----
    if (tid < 16) {
        float s = 0.f, s2 = 0.f;
        const float* hr = h + tid * 128;
        for (int c = 0; c < 128; ++c) { const float v = hr[c]; s += v; s2 += v * v; }
        const float m = s * (1.f / 128.f);
        const float var = s2 * (1.f / 128.f) - m * m;
        rmean[tid] = m;
        rrstd[tid] = rsqrtf(var + 1e-5f);
    }
    __syncthreads();

    // ---- apply LN * gamma + beta, then ReLU ----
    for (int i = tid; i < 16 * 128; i += 256) {
        const int r = i >> 7, c = i & 127;
        const float v = (h[i] - rmean[r]) * rrstd[r] * gam[c] + bet[c];
        h[i] = fmaxf(v, 0.f);
    }
    __syncthreads();

    // ---- GEMM2: waves 0..3, A from LDS, 128 -> 64 ----
    if (wave < 4) {  // wave-uniform branch: EXEC all-ones inside
        const int nbase = wave * 16;
        const int mn    = lane & 15;
        const int ks    = (lane >> 4) * 2;
        v8f acc = {};
        const float* hrow = h  + mn * 128;
        const float* wrow = W2 + (size_t)(nbase + mn) * 128;
        for (int k = 0; k < 128; k += 4) {
            v2f a, b;
            a.x = hrow[k + ks]; a.y = hrow[k + ks + 1];
            b.x = wrow[k + ks]; b.y = wrow[k + ks + 1];
            acc = __builtin_amdgcn_wmma_f32_16x16x4_f32(
                false, a, false, b, (short)0, acc, false, false);
        }
        const int mr = (lane >> 4) * 8;
        const int nc = nbase + (lane & 15);
#pragma unroll
        for (int r = 0; r < 8; ++r)
            o[(mr + r) * 64 + nc] = acc[r] + b2[nc];
    }
    __syncthreads();

    // ---- L2 norm per row ----
    if (tid < 16) {
        float s2 = 0.f;
        const float* orow = o + tid * 64;
        for (int c = 0; c < 64; ++c) { const float v = orow[c]; s2 += v * v; }
        rnorm[tid] = fmaxf(sqrtf(s2), 1e-12f);
    }
    __syncthreads();

    for (int i = tid; i < 16 * 64; i += 256) {
        const int r = i >> 6, c = i & 63;
        out[(size_t)(row0 + r) * 64 + c] = o[i] / rnorm[r];
    }
}

// =========================================================================
// Launch
// =========================================================================
extern "C" void kernel_launch(void* const* d_in, const int* in_sizes, int n_in,
                              void* d_out, int out_size, void* d_ws, size_t ws_size,
                              hipStream_t stream) {
    (void)n_in; (void)out_size; (void)ws_size;
    const int*   user_id     = (const int*)  d_in[0];
    const int*   history     = (const int*)  d_in[1];
    const int*   top_genres  = (const int*)  d_in[2];
    const int*   item_id     = (const int*)  d_in[3];
    const int*   tmdb_genres = (const int*)  d_in[4];
    const float* ts_diff     = (const float*)d_in[5];
    const float* u_avg       = (const float*)d_in[6];
    const float* activity    = (const float*)d_in[7];
    const float* rel_year    = (const float*)d_in[8];
    const float* i_avg       = (const float*)d_in[9];
    const float* revenue     = (const float*)d_in[10];
    const float* emb_item    = (const float*)d_in[11];
    const float* emb_genre   = (const float*)d_in[12];
    const float* emb_user    = (const float*)d_in[13];
    const float* ut_cont_w   = (const float*)d_in[14];
    const float* ut_cont_b   = (const float*)d_in[15];
    const float* ut_w1       = (const float*)d_in[16];
    const float* ut_b1       = (const float*)d_in[17];
    const float* ut_g        = (const float*)d_in[18];
    const float* ut_be       = (const float*)d_in[19];
    const float* ut_w2       = (const float*)d_in[20];
    const float* ut_b2       = (const float*)d_in[21];
    const float* it_cont_w   = (const float*)d_in[22];
    const float* it_cont_b   = (const float*)d_in[23];
    const float* it_w1       = (const float*)d_in[24];
    const float* it_b1       = (const float*)d_in[25];
    const float* it_g        = (const float*)d_in[26];
    const float* it_be       = (const float*)d_in[27];
    const float* it_w2       = (const float*)d_in[28];
    const float* it_b2       = (const float*)d_in[29];

    const int B = in_sizes[0];  // 16384

    float* u_cat = (float*)d_ws;                   // [B, 256]
    float* i_cat = u_cat + (size_t)B * 256;        // [B, 192]
    float* u_out = (float*)d_out;                  // [B, 64]
    float* i_out = u_out + (size_t)B * DIM;        // [B, 64]

    pool_features_kernel<<<(B + 7) / 8, 256, 0, stream>>>(
        user_id, history, top_genres, item_id, tmdb_genres, ts_diff,
        u_avg, activity, rel_year, i_avg, revenue,
        emb_item, emb_genre, emb_user,
        ut_cont_w, ut_cont_b, it_cont_w, it_cont_b,
        u_cat, i_cat, B);

    tower_mlp_kernel<256><<<B / 16, 256, 0, stream>>>(
        u_cat, ut_w1, ut_b1, ut_g, ut_be, ut_w2, ut_b2, u_out);

    tower_mlp_kernel<192><<<B / 16, 256, 0, stream>>>(
        i_cat, it_w1, it_b1, it_g, it_be, it_w2, it_b2, i_out);
}